// SymLoss_16003048145585
// MI455X (gfx1250) — compile-verified
//
#include <hip/hip_runtime.h>
#include <hip/hip_bf16.h>

#define LCONST 3.0f
#define BATCH  32
#define H_DIM  512
#define W_DIM  512
#define HW     (H_DIM * W_DIM)
#define CHUNKS 32                              // gridDim.x
#define BLOCK  256
#define TPS    (CHUNKS * BLOCK)                // 8192 threads per sample
#define QUADS  (HW / 4)                        // 65536 quads (4 pixels each)
#define ITERS  (QUADS / TPS)                   // 8 uniform iterations

__device__ __forceinline__ int imax(int a, int b) { return a > b ? a : b; }
__device__ __forceinline__ int imin(int a, int b) { return a < b ? a : b; }

// Full wave32 sum. On gfx1250 uses V_WMMA_F32_16X16X4_F32 as the reduction
// engine: A (16x4 f32, 2 VGPRs) holds per-lane partials in the K=0/K=2 slots,
// B = ones => D[m][n] = p[m] + p[m+16] replicated across columns. Summing the
// 8 D VGPRs gives the half-wave sums; one xor-16 shuffle completes it.
// Requires EXEC all-1s (call sites are fully uniform).
__device__ __forceinline__ float wave_reduce_sum(float v) {
#if defined(__gfx1250__)
    typedef __attribute__((ext_vector_type(2))) float v2f;
    typedef __attribute__((ext_vector_type(8))) float v8f;
    v2f a;    a[0] = v;       a[1] = 0.0f;
    v2f ones; ones[0] = 1.0f; ones[1] = 1.0f;
    v8f c = {};
    c = __builtin_amdgcn_wmma_f32_16x16x4_f32(
        /*neg_a=*/false, a, /*neg_b=*/false, ones,
        /*c_mod=*/(short)0, c, /*reuse_a=*/false, /*reuse_b=*/false);
    float t = ((c[0] + c[1]) + (c[2] + c[3])) + ((c[4] + c[5]) + (c[6] + c[7]));
    t += __shfl_xor(t, 16, 32);
    return t;
#else
    for (int off = 16; off > 0; off >>= 1) v += __shfl_xor(v, off, 32);
    return v;
#endif
}

__global__ void zero_acc_kernel(float* acc) {
    if (threadIdx.x < BATCH) acc[threadIdx.x] = 0.0f;
}

__global__ void __launch_bounds__(BLOCK)
sym_partial_kernel(const float* __restrict__ Flow,
                   const float* __restrict__ Asym,
                   const float* __restrict__ Bsym,
                   const float* __restrict__ Mask,
                   float* __restrict__ acc) {
    const int b = blockIdx.y;

    // Per-sample scalars (uniform -> SGPR)
    const float a0  = Asym[2 * b + 0];
    const float a1  = Asym[2 * b + 1];
    const float bs0 = Bsym[2 * b + 0];
    const float bs1 = Bsym[2 * b + 1];

    const float dxf = -LCONST * a0;
    const float dyf = fabsf(LCONST * a1);
    const float dy1 = floorf(dyf);
    const float dx1 = floorf(dxf);
    const float fy  = dyf - dy1;
    const float fx  = dxf - dx1;
    const int iy1 = (int)dy1;               // >= 0
    const int ix1 = (int)dx1;               // may be negative

    const int hp   = H_DIM - 1 - iy1;
    const int x_lo = imax(0, -ix1);
    const int x_hi = imin(W_DIM, W_DIM - 1 - ix1);

    const float w11 = (1.0f - fx) * (1.0f - fy);
    const float w12 = fx * (1.0f - fy);
    const float w21 = (1.0f - fx) * fy;
    const float w22 = fx * fy;

    const size_t plane = (size_t)HW;
    const float* __restrict__ f0 = Flow + ((size_t)b * 2 + 0) * plane;
    const float* __restrict__ f1 = Flow + ((size_t)b * 2 + 1) * plane;
    const float* __restrict__ m0 = Mask + ((size_t)b * 2 + 0) * plane;
    const float* __restrict__ m1 = Mask + ((size_t)b * 2 + 1) * plane;

    const int tid = blockIdx.x * BLOCK + threadIdx.x;   // 0..TPS-1

    float sum = 0.0f;

    for (int it = 0; it < ITERS; ++it) {                // compile-time uniform trip count
        const int q  = tid + it * TPS;                  // quad index, 0..QUADS-1
        const int y  = q >> 7;                          // 128 quads per row
        const int x0 = (q & 127) << 2;                  // x0 % 4 == 0 -> 16B aligned
        const int p  = y * W_DIM + x0;

        // Streamed data: one float4 per array (the only HBM traffic)
        const float4 s0 = *(const float4*)(f0 + p);
        const float4 s1 = *(const float4*)(f1 + p);
        const float4 q0 = *(const float4*)(m0 + p);
        const float4 q1 = *(const float4*)(m1 + p);

        // Tap rows (clamped identically to the reference; iy1 >= 0)
        const int gy1 = imin(y + iy1,     H_DIM - 1);
        const int gy2 = imin(y + iy1 + 1, H_DIM - 1);
        const float* __restrict__ r0a = f0 + gy1 * W_DIM;
        const float* __restrict__ r0b = f0 + gy2 * W_DIM;
        const float* __restrict__ r1a = f1 + gy1 * W_DIM;
        const float* __restrict__ r1b = f1 + gy2 * W_DIM;

        // 5-wide overlapping spans: t[j] = gx1 of pixel j, t[j+1] = gx2 of pixel j
        float t0a[5], t0b[5], t1a[5], t1b[5];
        #pragma unroll
        for (int j = 0; j < 5; ++j) {
            const int gx = imin(imax(x0 + j + ix1, 0), W_DIM - 1);
            t0a[j] = r0a[gx];
            t0b[j] = r0b[gx];
            t1a[j] = r1a[gx];
            t1b[j] = r1b[gx];
        }

        const float sf0[4] = { s0.x, s0.y, s0.z, s0.w };
        const float sf1[4] = { s1.x, s1.y, s1.z, s1.w };
        const float sm[4]  = { q0.x + q1.x, q0.y + q1.y, q0.z + q1.z, q0.w + q1.w };

        const bool yok = (y < hp);
        #pragma unroll
        for (int j = 0; j < 4; ++j) {
            const int x = x0 + j;
            const bool valid = yok & (x >= x_lo) & (x < x_hi);

            const float h0 = w11 * t0a[j] + w12 * t0a[j + 1]
                           + w21 * t0b[j] + w22 * t0b[j + 1];
            const float h1 = w11 * t1a[j] + w12 * t1a[j + 1]
                           + w21 * t1b[j] + w22 * t1b[j + 1];

            const float d0 = sf0[j] - h0;
            const float d1 = sf1[j] - h1;
            float s = d1 * bs0 - d0 * bs1;
            s = s * s;

            sum = fmaf(valid ? sm[j] : 0.0f, s, sum);
        }
    }

    // EXEC all-1s (fully uniform control flow above) -> WMMA reduction legal
    const float wsum = wave_reduce_sum(sum);

    __shared__ float lds_wsum[BLOCK / 32];
    const int lane = threadIdx.x & 31;
    const int wave = threadIdx.x >> 5;
    if (lane == 0) lds_wsum[wave] = wsum;
    __syncthreads();
    if (threadIdx.x == 0) {
        float bt = 0.0f;
        #pragma unroll
        for (int i = 0; i < BLOCK / 32; ++i) bt += lds_wsum[i];
        atomicAdd(&acc[b], bt);
    }
}

__global__ void finalize_kernel(const float* __restrict__ acc,
                                const float* __restrict__ Asym,
                                float* __restrict__ out) {
    const int b = threadIdx.x;   // exactly 32 threads = 1 wave, EXEC all-1s
    const float a0 = Asym[2 * b + 0];
    const float a1 = Asym[2 * b + 1];
    const float dxf = -LCONST * a0;
    const float dyf = fabsf(LCONST * a1);
    const int iy1 = (int)floorf(dyf);
    const int ix1 = (int)floorf(dxf);
    const int hp   = H_DIM - 1 - iy1;
    const int x_lo = imax(0, -ix1);
    const int x_hi = imin(W_DIM, W_DIM - 1 - ix1);

    const float denom = 2.0f * (float)imax(hp, 1) * (float)imax(x_hi - x_lo, 1);
    float v = acc[b] / denom;

    v = wave_reduce_sum(v);
    if (b == 0) out[0] = v * (1.0f / (float)BATCH);
}

extern "C" void kernel_launch(void* const* d_in, const int* in_sizes, int n_in,
                              void* d_out, int out_size, void* d_ws, size_t ws_size,
                              hipStream_t stream) {
    const float* Flow = (const float*)d_in[0];   // (32,2,512,512) f32
    const float* Asym = (const float*)d_in[1];   // (32,2) f32
    const float* Bsym = (const float*)d_in[2];   // (32,2) f32
    const float* Mask = (const float*)d_in[3];   // (32,2,512,512) f32
    float* out = (float*)d_out;                  // scalar f32
    float* acc = (float*)d_ws;                   // 32 floats of scratch

    zero_acc_kernel<<<1, 32, 0, stream>>>(acc);

    dim3 grid(CHUNKS, BATCH);
    sym_partial_kernel<<<grid, BLOCK, 0, stream>>>(Flow, Asym, Bsym, Mask, acc);

    finalize_kernel<<<1, 32, 0, stream>>>(acc, Asym, out);
}